// EnsembleGCN_63642825392598
// MI455X (gfx1250) — compile-verified
//
#include <hip/hip_runtime.h>
#include <hip/hip_bf16.h>

typedef float v2f __attribute__((ext_vector_type(2)));
typedef float v8f __attribute__((ext_vector_type(8)));

namespace {
constexpr int NN   = 600;   // nodes
constexpr int CCLS = 5;     // classes
constexpr int NQ   = 80;    // query nodes (last rows)
constexpr int NS   = NN - NQ;   // 520 support nodes
constexpr int FE   = 256;   // final embed
constexpr int CATD = 261;   // 128+128+5
constexpr int CATP = 272;   // padded stride (mult of 16)
constexpr int FCW  = 510;   // 256 + 254
constexpr int FCP  = 512;   // padded stride
constexpr int KC   = 3;     // conv1d kernel

constexpr int KT   = 16;    // K chunk staged per iteration
constexpr int BN   = 128;   // N columns per block (8 waves x 16)
constexpr int ASTR = 17;    // A tile LDS stride (conflict-free: 17 coprime 64)
constexpr int BSTR = 136;   // B tile LDS stride (row+2 -> bank+16, disjoint)
}

// ---------------------------------------------------------------------------
// fp32 GEMM: C[M x Nw] = A[M x K] * W[K x Nw] (+ bias), LDS-staged tiles.
// Block = 256 threads (8 waves); block computes a 16(M) x 128(N) strip,
// wave w owns the 16x16 tile at columns nb0 + 16w.  Per 16-deep K chunk:
// cooperative fill of As(16x16)/Bs(16x128) -- unguarded float4 fast path for
// interior chunks, guarded scalar path for edges -- then 4 guard-free
// V_WMMA_F32_16X16X4_F32 per wave out of LDS.
// ---------------------------------------------------------------------------
__global__ __launch_bounds__(256)
void gemm_wmma_f32(const float* __restrict__ A, int lda,
                   const float* __restrict__ W, int ldw, int K,
                   const float* __restrict__ bias,
                   float* __restrict__ Cout, int ldc,
                   int M, int Nw) {
  __shared__ float As[KT * ASTR];   // [m within tile][k within chunk]
  __shared__ float Bs[KT * BSTR];   // [k within chunk][col within strip]

  const int tid  = threadIdx.x;
  const int lane = tid & 31;
  const int wave = tid >> 5;         // 0..7
  const int l16  = lane & 15;
  const int hi   = lane >> 4;        // 0: k-pair {0,1}, 1: k-pair {2,3}
  const int m0   = blockIdx.x * 16;
  const int nb0  = blockIdx.y * BN;
  const int col  = nb0 + wave * 16 + l16;   // this lane's output column

  // cooperative fill indices
  const int ar = tid >> 4;           // 0..15  A row within tile
  const int ak = tid & 15;           // 0..15  A k within chunk
  const bool arow_ok = (m0 + ar) < M;
  const float* Arow = A + (size_t)(m0 + ar) * lda;

  // fast-path eligibility (uniform across block)
  const bool full_m = (m0 + 16) <= M;
  const bool full_n = ((nb0 + BN) <= Nw) && ((ldw & 3) == 0);
  // B float4 fill indices: thread -> (row kk, col group), 2 rows apart by 8
  const int bk = tid >> 5;           // 0..7
  const int bc = (tid & 31) * 4;     // 0,4,...,124

  v8f acc = {};
  for (int k = 0; k < K; k += KT) {
    __syncthreads();   // previous chunk fully consumed
    const bool full_k = (k + KT) <= K;
    if (full_k && full_m && full_n) {
      // ---- unguarded, wide fill ----
      As[ar * ASTR + ak] = Arow[k + ak];
      const float* Wk = W + (size_t)k * ldw + nb0;
      float4 b0 = *(const float4*)(Wk + bk * ldw + bc);
      float4 b1 = *(const float4*)(Wk + (bk + 8) * ldw + bc);
      float* B0 = &Bs[bk * BSTR + bc];
      B0[0] = b0.x; B0[1] = b0.y; B0[2] = b0.z; B0[3] = b0.w;
      float* B1 = &Bs[(bk + 8) * BSTR + bc];
      B1[0] = b1.x; B1[1] = b1.y; B1[2] = b1.z; B1[3] = b1.w;
    } else {
      // ---- guarded scalar fill (M/K/N edges, tiny W_lab GEMM) ----
      As[ar * ASTR + ak] = (arow_ok && (k + ak) < K) ? Arow[k + ak] : 0.f;
#pragma unroll
      for (int q = 0; q < 8; ++q) {
        const int e  = tid + 256 * q;
        const int kk = e >> 7;          // 0..15
        const int c  = e & 127;         // 0..127
        Bs[kk * BSTR + c] = ((k + kk) < K && (nb0 + c) < Nw)
                                ? W[(size_t)(k + kk) * ldw + nb0 + c] : 0.f;
      }
    }
    // prefetch next A chunk (global_prefetch_b8)
    if (arow_ok && (k + KT + ak) < K) __builtin_prefetch(&Arow[k + KT + ak], 0, 1);
    __syncthreads();
    // 4 guard-free WMMAs out of LDS
#pragma unroll
    for (int kk = 0; kk < KT; kk += 4) {
      const int ka = kk + 2 * hi;
      v2f a, b;
      a.x = As[l16 * ASTR + ka];
      a.y = As[l16 * ASTR + ka + 1];
      b.x = Bs[ka * BSTR + wave * 16 + l16];
      b.y = Bs[(ka + 1) * BSTR + wave * 16 + l16];
      acc = __builtin_amdgcn_wmma_f32_16x16x4_f32(false, a, false, b,
                                                  (short)0, acc, false, false);
    }
  }

  float bv = 0.f;
  if (bias != nullptr && col < Nw) bv = bias[col];
  if (full_m && (nb0 + BN) <= Nw) {
#pragma unroll
    for (int r = 0; r < 8; ++r)
      Cout[(size_t)(m0 + r + 8 * hi) * ldc + col] = acc[r] + bv;
  } else {
#pragma unroll
    for (int r = 0; r < 8; ++r) {
      const int rr = m0 + r + 8 * hi;
      if (rr < M && col < Nw) Cout[(size_t)rr * ldc + col] = acc[r] + bv;
    }
  }
}

// ---------------------------------------------------------------------------
// Degree counting for GCN normalization (dst in-degree; +1 self loop later)
// ---------------------------------------------------------------------------
__global__ void k_deg(const int* __restrict__ in_dst, int Ein,
                      const int* __restrict__ fin_dst, int Efin,
                      float* __restrict__ deg_in, float* __restrict__ deg_fin) {
  int t = blockIdx.x * blockDim.x + threadIdx.x;
  if (t < Ein)  atomicAdd(&deg_in[in_dst[t]], 1.f);
  if (t < Efin) atomicAdd(&deg_fin[fin_dst[t]], 1.f);
}

__global__ void k_dinv(float* __restrict__ deg_in, float* __restrict__ deg_fin, int n) {
  int i = blockIdx.x * blockDim.x + threadIdx.x;
  if (i < n) {
    deg_in[i]  = rsqrtf(deg_in[i]  + 1.f);   // +1 = self loop
    deg_fin[i] = rsqrtf(deg_fin[i] + 1.f);
  }
}

// ---------------------------------------------------------------------------
// Normalized edge scatter: acc[d][f] += dinv[s]*dinv[d]*h[s][f]
// ---------------------------------------------------------------------------
__global__ void agg_edges(const float* __restrict__ h, int ldh,
                          float* __restrict__ acc, int ldacc,
                          const int* __restrict__ src, const int* __restrict__ dst,
                          int E, int F, const float* __restrict__ dinv) {
  long long total = (long long)E * F;
  for (long long t = (long long)blockIdx.x * blockDim.x + threadIdx.x; t < total;
       t += (long long)gridDim.x * blockDim.x) {
    int e = (int)(t / F);
    int f = (int)(t - (long long)e * F);
    int s = src[e], d = dst[e];
    atomicAdd(&acc[(size_t)d * ldacc + f], dinv[s] * dinv[d] * h[(size_t)s * ldh + f]);
  }
}

// out[i][f] = act(acc[i][f] + dinv[i]^2 * h[i][f] + bias[f]); slope=1 -> identity
__global__ void gcn_finalize(const float* __restrict__ acc, int ldacc,
                             const float* __restrict__ h, int ldh,
                             const float* __restrict__ dinv,
                             const float* __restrict__ bias,
                             float* __restrict__ out, int ldo,
                             int M, int F, float slope) {
  int t = blockIdx.x * blockDim.x + threadIdx.x;
  if (t >= M * F) return;
  int i = t / F, f = t - i * F;
  float di = dinv[i];
  float v = acc[(size_t)i * ldacc + f] + di * di * h[(size_t)i * ldh + f] + bias[f];
  out[(size_t)i * ldo + f] = v > 0.f ? v : slope * v;
}

// one-hot / uniform prior columns of cated
__global__ void k_onehot(const int* __restrict__ labels, float* __restrict__ cated) {
  int i = blockIdx.x * blockDim.x + threadIdx.x;
  if (i >= NN) return;
#pragma unroll
  for (int c = 0; c < CCLS; ++c)
    cated[(size_t)i * CATP + 2 * 128 + c] = (i < NS) ? (labels[i] == c ? 1.f : 0.f)
                                                     : (1.f / CCLS);
}

// ---------------------------------------------------------------------------
// Dense GATv2 (cross-class complete structure) + ELU + conv1d + sigmoid,
// fused. One 256-thread block per destination node i.
// Edge (j -> i) exists iff pseudo-label(j) != pseudo-label(i), plus self loop.
// Softmax computed without max-shift (shift-invariant; logits are tiny).
// ---------------------------------------------------------------------------
__device__ __forceinline__ int pseudo_label(const int* labels, int i) {
  return (i < NS) ? labels[i] : -(i - NS + 1);
}

__global__ __launch_bounds__(256)
void gat_conv_kernel(const float* __restrict__ hl, const float* __restrict__ hr,
                     const float* __restrict__ att, const float* __restrict__ bgat,
                     const int* __restrict__ labels,
                     const float* __restrict__ convw, const float* __restrict__ convb,
                     float* __restrict__ fc) {
  const int i    = blockIdx.x;     // destination node
  const int tid  = threadIdx.x;    // 0..255
  const int lane = tid & 31;
  const int wave = tid >> 5;       // 0..7

  __shared__ float sAcc[FE];
  __shared__ float sSum;
  if (tid < FE) sAcc[tid] = 0.f;
  if (tid == 0) sSum = 0.f;
  __syncthreads();

  float hrr[8], aa[8], acc[8];
#pragma unroll
  for (int r = 0; r < 8; ++r) {
    int f = lane + 32 * r;
    hrr[r] = hr[(size_t)i * FE + f];
    aa[r]  = att[f];
    acc[r] = 0.f;
  }
  float sumex = 0.f;
  const int ti = pseudo_label(labels, i);

  for (int j = wave; j < NN; j += 8) {
    const int tj = pseudo_label(labels, j);
    if (j != i && tj == ti) continue;   // not a cross edge, not self loop
    float x[8], p = 0.f;
#pragma unroll
    for (int r = 0; r < 8; ++r) {
      x[r] = hl[(size_t)j * FE + lane + 32 * r];
      float s = x[r] + hrr[r];
      float lr = s > 0.f ? s : 0.2f * s;   // GATv2 leaky slope
      p += lr * aa[r];
    }
#pragma unroll
    for (int off = 16; off > 0; off >>= 1) p += __shfl_xor(p, off, 32);
    float ex = __expf(p);
    sumex += ex;
#pragma unroll
    for (int r = 0; r < 8; ++r) acc[r] += ex * x[r];
  }
#pragma unroll
  for (int r = 0; r < 8; ++r) atomicAdd(&sAcc[lane + 32 * r], acc[r]);
  if (lane == 0) atomicAdd(&sSum, sumex);
  __syncthreads();

  if (tid < FE) {
    float v = sAcc[tid] / sSum + bgat[tid];
    sAcc[tid] = v > 0.f ? v : (__expf(v) - 1.f);   // ELU
  }
  __syncthreads();

  if (tid < FE - KC + 1) {   // conv1d (correlation) + sigmoid -> fc[:, 256:510]
    float c = convw[0] * sAcc[tid] + convw[1] * sAcc[tid + 1] +
              convw[2] * sAcc[tid + 2] + convb[0];
    fc[(size_t)i * FCP + FE + tid] = 1.f / (1.f + __expf(-c));
  }
}

// ---------------------------------------------------------------------------
extern "C" void kernel_launch(void* const* d_in, const int* in_sizes, int n_in,
                              void* d_out, int out_size, void* d_ws, size_t ws_size,
                              hipStream_t stream) {
  const float* f0      = (const float*)d_in[0];
  const float* f1      = (const float*)d_in[1];
  const int*   labels  = (const int*)  d_in[2];
  const int*   in_src  = (const int*)  d_in[3];
  const int*   in_dst  = (const int*)  d_in[4];
  // d_in[5], d_in[6] (cross edges) replaced by dense pseudo-label predicate
  const int*   fin_src = (const int*)  d_in[7];
  const int*   fin_dst = (const int*)  d_in[8];
  const float* W_f0 = (const float*)d_in[9];   const float* b_f0 = (const float*)d_in[10];
  const float* W_f1 = (const float*)d_in[11];  const float* b_f1 = (const float*)d_in[12];
  const float* W_in = (const float*)d_in[13];  const float* b_in = (const float*)d_in[14];
  const float* Wl   = (const float*)d_in[15];  const float* bl   = (const float*)d_in[16];
  const float* Wr   = (const float*)d_in[17];  const float* br   = (const float*)d_in[18];
  const float* att  = (const float*)d_in[19];  const float* bgat = (const float*)d_in[20];
  const float* cw   = (const float*)d_in[21];  const float* cb   = (const float*)d_in[22];
  const float* Wlab = (const float*)d_in[23];  const float* blab = (const float*)d_in[24];
  float* out = (float*)d_out;

  const int Ein  = in_sizes[3];
  const int Efin = in_sizes[7];

  // ---- workspace layout (all float, zeroed each launch) ----
  float* ws = (float*)d_ws;
  size_t off = 0;
  auto alloc = [&](size_t n) { float* p = ws + off; off += n; return p; };
  float* dinv_in  = alloc(NN);
  float* dinv_fin = alloc(NN);
  float* h0x      = alloc((size_t)NN * 128);
  float* h1x      = alloc((size_t)NN * 128);
  float* acc0     = alloc((size_t)NN * 128);
  float* acc1     = alloc((size_t)NN * 128);
  float* cated    = alloc((size_t)NN * CATP);
  float* gcnin    = alloc((size_t)NN * FE);
  float* gcnacc   = alloc((size_t)NN * FE);
  float* hl       = alloc((size_t)NN * FE);
  float* hr       = alloc((size_t)NN * FE);
  float* fc       = alloc((size_t)NN * FCP);
  float* xwlab    = alloc((size_t)NN * 8);
  float* outacc   = alloc((size_t)NN * 8);
  if (off * sizeof(float) > ws_size) return;  // insufficient scratch (shouldn't happen)

  hipMemsetAsync(d_ws, 0, off * sizeof(float), stream);

  // degrees -> dinv
  {
    int emax = Ein > Efin ? Ein : Efin;
    k_deg<<<(emax + 255) / 256, 256, 0, stream>>>(in_dst, Ein, fin_dst, Efin,
                                                  dinv_in, dinv_fin);
    k_dinv<<<(NN + 255) / 256, 256, 0, stream>>>(dinv_in, dinv_fin, NN);
  }

  const int mT = (NN + 15) / 16;   // 38 row tiles

  // features @ W_f{0,1}   (Nw = 128 -> one 128-col strip)
  gemm_wmma_f32<<<dim3(mT, 1), 256, 0, stream>>>(f0, 1024, W_f0, 128, 1024,
                                                 nullptr, h0x, 128, NN, 128);
  gemm_wmma_f32<<<dim3(mT, 1), 256, 0, stream>>>(f1, 1024, W_f1, 128, 1024,
                                                 nullptr, h1x, 128, NN, 128);
  // in-class GCN aggregation -> leaky_relu -> cated columns
  {
    long long tot = (long long)Ein * 128;
    int blk = (int)((tot + 255) / 256);
    agg_edges<<<blk, 256, 0, stream>>>(h0x, 128, acc0, 128, in_src, in_dst, Ein, 128, dinv_in);
    agg_edges<<<blk, 256, 0, stream>>>(h1x, 128, acc1, 128, in_src, in_dst, Ein, 128, dinv_in);
    int ft = (NN * 128 + 255) / 256;
    gcn_finalize<<<ft, 256, 0, stream>>>(acc0, 128, h0x, 128, dinv_in, b_f0,
                                         cated, CATP, NN, 128, 0.01f);
    gcn_finalize<<<ft, 256, 0, stream>>>(acc1, 128, h1x, 128, dinv_in, b_f1,
                                         cated + 128, CATP, NN, 128, 0.01f);
  }
  k_onehot<<<(NN + 255) / 256, 256, 0, stream>>>(labels, cated);

  // cated @ {W_in, Wl, Wr}   (Nw = 256 -> two 128-col strips)
  gemm_wmma_f32<<<dim3(mT, 2), 256, 0, stream>>>(cated, CATP, W_in, FE, CATD,
                                                 nullptr, gcnin, FE, NN, FE);
  gemm_wmma_f32<<<dim3(mT, 2), 256, 0, stream>>>(cated, CATP, Wl, FE, CATD,
                                                 bl, hl, FE, NN, FE);
  gemm_wmma_f32<<<dim3(mT, 2), 256, 0, stream>>>(cated, CATP, Wr, FE, CATD,
                                                 br, hr, FE, NN, FE);
  // gcn_emb -> fc[:, 0:256]
  {
    long long tot = (long long)Ein * FE;
    agg_edges<<<(int)((tot + 255) / 256), 256, 0, stream>>>(gcnin, FE, gcnacc, FE,
                                                            in_src, in_dst, Ein, FE, dinv_in);
    gcn_finalize<<<(NN * FE + 255) / 256, 256, 0, stream>>>(gcnacc, FE, gcnin, FE,
                                                            dinv_in, b_in, fc, FCP,
                                                            NN, FE, 0.01f);
  }
  // GATv2 + ELU + conv1d + sigmoid -> fc[:, 256:510]
  gat_conv_kernel<<<NN, 256, 0, stream>>>(hl, hr, att, bgat, labels, cw, cb, fc);

  // fc @ W_lab, then query<->support pooling GCN -> output
  gemm_wmma_f32<<<dim3(mT, 1), 256, 0, stream>>>(fc, FCP, Wlab, CCLS, FCW,
                                                 nullptr, xwlab, 8, NN, CCLS);
  {
    long long tot = (long long)Efin * CCLS;
    agg_edges<<<(int)((tot + 255) / 256), 256, 0, stream>>>(xwlab, 8, outacc, 8,
                                                            fin_src, fin_dst, Efin,
                                                            CCLS, dinv_fin);
    gcn_finalize<<<(NN * CCLS + 255) / 256, 256, 0, stream>>>(outacc, 8, xwlab, 8,
                                                              dinv_fin, blab, out, CCLS,
                                                              NN, CCLS, 1.0f);
  }
}